// HyperbolicGraphConvolution_35476429864966
// MI455X (gfx1250) — compile-verified
//
#include <hip/hip_runtime.h>
#include <math.h>

typedef float v2f __attribute__((ext_vector_type(2)));
typedef float v8f __attribute__((ext_vector_type(8)));

#define MIN_NORM 1e-15f
#define BALL_EPS 4e-3f

__device__ __forceinline__ float artanh_f(float x) {
    x = fminf(fmaxf(x, -1.0f + 1e-7f), 1.0f - 1e-7f);
    return 0.5f * (log1pf(x) - log1pf(-x));
}

// Combined expmap0 (c=1) followed by proj: result = scale * u, given |u|^2.
__device__ __forceinline__ float expmap0_proj_scale(float u2) {
    float tn = sqrtf(u2);
    float un = fmaxf(tn, MIN_NORM);
    float th = tanhf(un);
    float s  = th / un;                       // e = s*u
    float en = fmaxf(s * tn, MIN_NORM);       // |e| (clamped)
    float maxn = 1.0f - BALL_EPS;
    float f  = en > maxn ? maxn / en : 1.0f;  // proj factor
    return s * f;
}

// ---------------------------------------------------------------- zero scratch
__global__ void zero_kernel(float4* p, size_t n4) {
    size_t i = (size_t)blockIdx.x * blockDim.x + threadIdx.x;
    if (i < n4) p[i] = make_float4(0.f, 0.f, 0.f, 0.f);
}

__global__ void select_init_kernel(unsigned* state, unsigned k) {
    state[0] = 0u;   // prefix
    state[1] = k;    // remaining
    state[2] = 0u;   // T bits
}

// ------------------------------------------------------- hyp_bias = proj(expmap0(bias))
__global__ void bias_kernel(const float* __restrict__ bias, float* __restrict__ hb) {
    __shared__ float s[64];
    __shared__ float s_n2;
    int t = threadIdx.x;                 // 64 threads
    float b = bias[t];
    s[t] = b * b;
    __syncthreads();
    if (t == 0) {
        float a = 0.f;
        for (int i = 0; i < 64; ++i) a += s[i];
        s_n2 = a;
    }
    __syncthreads();
    float n2 = s_n2;
    float g = expmap0_proj_scale(n2);    // hb = g * bias
    hb[t] = g * b;
    if (t == 0) hb[64] = g * g * n2;     // |hyp_bias|^2
}

// -------------------------------------------- HypLinear + logmap0 via f32 WMMA
// One wave: 16 nodes x 64 outputs via 16 K-steps x 4 N-tiles of
// V_WMMA_F32_16X16X4_F32. Block = 8 waves = 128 nodes.
__global__ __launch_bounds__(256) void hyplinear_kernel(
        const float* __restrict__ x, const float* __restrict__ W,
        const float* __restrict__ hb, float* __restrict__ xtan, int N)
{
    __shared__ float lds_mx[128 * 68];   // stride 68 avoids bank conflicts
    __shared__ float lds_hb[64];
    int tid = threadIdx.x;
    if (tid < 64) lds_hb[tid] = hb[tid];

    int wave  = tid >> 5;
    int lane  = tid & 31;
    int row16 = lane & 15;               // M (A) / N (B,C,D) position
    int hi    = lane >> 4;               // which K half / which M half for C
    int nodeBase = blockIdx.x * 128 + wave * 16;
    int arow = nodeBase + row16;
    if (arow > N - 1) arow = N - 1;      // clamp loads; stores guarded later
    const float* xr = x + (size_t)arow * 64;

    v8f c0 = {}, c1 = {}, c2 = {}, c3 = {};
#pragma unroll
    for (int k = 0; k < 16; ++k) {
        int kb = k * 4 + hi * 2;
        // A (16x4): lane row16 holds x[row, kb], x[row, kb+1]
        v2f a; a.x = xr[kb]; a.y = xr[kb + 1];
        // B (4x16): B[k,n] = W[n_out, k] (x @ W^T)
        v2f b0, b1, b2, b3;
        b0.x = W[(row16)      * 64 + kb]; b0.y = W[(row16)      * 64 + kb + 1];
        b1.x = W[(16 + row16) * 64 + kb]; b1.y = W[(16 + row16) * 64 + kb + 1];
        b2.x = W[(32 + row16) * 64 + kb]; b2.y = W[(32 + row16) * 64 + kb + 1];
        b3.x = W[(48 + row16) * 64 + kb]; b3.y = W[(48 + row16) * 64 + kb + 1];
        c0 = __builtin_amdgcn_wmma_f32_16x16x4_f32(false, a, false, b0, (short)0, c0, false, false);
        c1 = __builtin_amdgcn_wmma_f32_16x16x4_f32(false, a, false, b1, (short)0, c1, false, false);
        c2 = __builtin_amdgcn_wmma_f32_16x16x4_f32(false, a, false, b2, (short)0, c2, false, false);
        c3 = __builtin_amdgcn_wmma_f32_16x16x4_f32(false, a, false, b3, (short)0, c3, false, false);
    }
    // C/D layout: VGPR r, lanes [0,15]: M=r, N=lane; lanes [16,31]: M=r+8
    int lbase = wave * 16 + hi * 8;
#pragma unroll
    for (int r = 0; r < 8; ++r) {
        lds_mx[(lbase + r) * 68 +      row16] = c0[r];
        lds_mx[(lbase + r) * 68 + 16 + row16] = c1[r];
        lds_mx[(lbase + r) * 68 + 32 + row16] = c2[r];
        lds_mx[(lbase + r) * 68 + 48 + row16] = c3[r];
    }
    __syncthreads();

    // Per-node epilogue: mobius_matvec tail + proj + mobius_add + proj + logmap0
    if (tid < 128) {
        int node = blockIdx.x * 128 + tid;
        if (node < N) {
            const float* xrow = x + (size_t)node * 64;
            const float* m = &lds_mx[tid * 68];
            float xn2 = 0.f, mxn2 = 0.f, dotmb = 0.f;
            for (int d = 0; d < 64; ++d) {
                float xv = xrow[d]; xn2 = fmaf(xv, xv, xn2);
                float mv = m[d];    mxn2 = fmaf(mv, mv, mxn2);
                dotmb = fmaf(mv, lds_hb[d], dotmb);
            }
            float tn_x = sqrtf(xn2);
            float xn   = fmaxf(tn_x, MIN_NORM);
            float tn_m = sqrtf(mxn2);
            float mxn  = fmaxf(tn_m, MIN_NORM);
            float th   = tanhf(mxn / xn * artanh_f(xn));     // sc = 1
            float s_res = th / mxn;                          // res = s_res*mx
            if (mxn2 == 0.0f) s_res = 0.0f;
            float maxn = 1.0f - BALL_EPS;
            float resn = fmaxf(fabsf(s_res) * tn_m, MIN_NORM);
            float f1   = resn > maxn ? maxn / resn : 1.0f;
            float s_h  = s_res * f1;                         // h = s_h*mx
            float x2 = s_h * s_h * mxn2;
            float xy = s_h * dotmb;
            float y2 = hb[64];
            float ca = 1.0f + 2.0f * xy + y2;
            float cb = 1.0f - x2;
            float den = fmaxf(1.0f + 2.0f * xy + x2 * y2, MIN_NORM);
            float A1 = ca * s_h / den;                       // add = A1*mx + A2*hb
            float A2 = cb / den;
            float addn2 = A1 * A1 * mxn2 + 2.0f * A1 * A2 * dotmb + A2 * A2 * y2;
            float tn_a = sqrtf(fmaxf(addn2, 0.0f));
            float an   = fmaxf(tn_a, MIN_NORM);
            float f2   = an > maxn ? maxn / an : 1.0f;       // proj
            float pn   = fmaxf(f2 * tn_a, MIN_NORM);
            float lcoef = artanh_f(pn) / pn;                 // logmap0
            float K1 = lcoef * f2 * A1;
            float K2 = lcoef * f2 * A2;
            float* o = xtan + (size_t)node * 64;
            for (int d = 0; d < 64; ++d)
                o[d] = fmaf(K1, m[d], K2 * lds_hb[d]);
        }
    }
}

// ------------------------------------------------------------------ edge passes
__global__ void deg_kernel(const int* __restrict__ ei, float* __restrict__ deg, int E) {
    int e = blockIdx.x * blockDim.x + threadIdx.x;
    if (e >= E) return;
    int src = ei[e], dst = ei[E + e];
    if (src != dst) atomicAdd(&deg[src], 1.0f);
}

__global__ void dinv_kernel(const float* __restrict__ deg, float* __restrict__ dinv, int N) {
    int n = blockIdx.x * blockDim.x + threadIdx.x;
    if (n >= N) return;
    float d = deg[n];
    dinv[n] = d > 0.f ? 1.0f / sqrtf(d) : 0.f;
}

// sum_neigh[dst] += x_tan[src]; acc_info[dst] += norm_w * x_tan[src]
__global__ void edge_passA(const int* __restrict__ ei, const float* __restrict__ xtan,
                           const float* __restrict__ dinv, float* __restrict__ neigh,
                           float* __restrict__ info, int E)
{
    int gid = blockIdx.x * blockDim.x + threadIdx.x;
    int e = gid >> 4;
    if (e >= E) return;
    int d0 = (gid & 15) << 2;
    int src = ei[e], dst = ei[E + e];
    const float4 v = *(const float4*)(xtan + (size_t)src * 64 + d0);
    float* np = neigh + (size_t)dst * 64 + d0;
    atomicAdd(np + 0, v.x); atomicAdd(np + 1, v.y);
    atomicAdd(np + 2, v.z); atomicAdd(np + 3, v.w);
    if (src != dst) {
        float nw = -(dinv[src] * dinv[dst]);
        float* ip = info + (size_t)dst * 64 + d0;
        atomicAdd(ip + 0, nw * v.x); atomicAdd(ip + 1, nw * v.y);
        atomicAdd(ip + 2, nw * v.z); atomicAdd(ip + 3, nw * v.w);
    }
}

__global__ void score_kernel(const float* __restrict__ xtan, const float* __restrict__ info,
                             float* __restrict__ score, int N)
{
    int n = blockIdx.x * blockDim.x + threadIdx.x;
    if (n >= N) return;
    const float* a = xtan + (size_t)n * 64;
    const float* b = info + (size_t)n * 64;
    float s = 0.f;
    for (int d = 0; d < 64; ++d) s += fabsf(a[d] + b[d]);
    score[n] = s;
}

// ------------------------------------------- device-side radix top-k threshold
// scores >= 0 so IEEE bits are order-preserving as uint32.
__global__ void hist_kernel(const float* __restrict__ score, unsigned* __restrict__ hist,
                            const unsigned* __restrict__ state, int pass, int N)
{
    int n = blockIdx.x * blockDim.x + threadIdx.x;
    if (n >= N) return;
    unsigned bits = __float_as_uint(score[n]);
    bool match = (pass == 0) || ((bits >> (32 - 8 * pass)) == state[0]);
    if (match) atomicAdd(&hist[(bits >> (24 - 8 * pass)) & 255u], 1u);
}

__global__ void scan_kernel(unsigned* __restrict__ hist, unsigned* __restrict__ state, int pass) {
    __shared__ unsigned h[256];
    int t = threadIdx.x;
    h[t] = hist[t];
    __syncthreads();
    if (t == 0) {
        unsigned remaining = state[1];
        unsigned cum = 0, bin = 0;
        for (int b = 255; b >= 0; --b) {
            unsigned c = h[b];
            if (cum + c >= remaining) { bin = (unsigned)b; break; }
            cum += c;
        }
        state[0] = (state[0] << 8) | bin;
        state[1] = remaining - cum;
        if (pass == 3) state[2] = state[0];   // bits of k-th largest -> T
    }
    __syncthreads();
    hist[t] = 0;                              // ready for next pass
}

// sum_sel[dst] += sel[src]*x_tan[src]   (sel computed on the fly: score > T)
__global__ void edge_passC(const int* __restrict__ ei, const float* __restrict__ xtan,
                           const float* __restrict__ score, const unsigned* __restrict__ state,
                           float* __restrict__ selsum, int E)
{
    int gid = blockIdx.x * blockDim.x + threadIdx.x;
    int e = gid >> 4;
    if (e >= E) return;
    int src = ei[e], dst = ei[E + e];
    float T = __uint_as_float(state[2]);
    if (!(score[src] > T)) return;
    int d0 = (gid & 15) << 2;
    const float4 v = *(const float4*)(xtan + (size_t)src * 64 + d0);
    float* p = selsum + (size_t)dst * 64 + d0;
    atomicAdd(p + 0, v.x); atomicAdd(p + 1, v.y);
    atomicAdd(p + 2, v.z); atomicAdd(p + 3, v.w);
}

__global__ void gate_kernel(const float* __restrict__ selsum, const float* __restrict__ neigh,
                            const float* __restrict__ lw_w, const float* __restrict__ lw_b,
                            float* __restrict__ gate, int N)
{
    int n = blockIdx.x * blockDim.x + threadIdx.x;
    if (n >= N) return;
    const float* a = selsum + (size_t)n * 64;
    const float* b = neigh + (size_t)n * 64;
    float acc = lw_b[0];
    for (int d = 0; d < 64; ++d)
        acc = fmaf(lw_w[d], a[d], fmaf(lw_w[64 + d], b[d], acc));
    gate[n] = 1.0f / (1.0f + expf(-acc));
}

// acc_g[dst] += gate[src]*sel[src]*x_tan[src]
__global__ void edge_passD(const int* __restrict__ ei, const float* __restrict__ xtan,
                           const float* __restrict__ score, const float* __restrict__ gate,
                           const unsigned* __restrict__ state, float* __restrict__ accg, int E)
{
    int gid = blockIdx.x * blockDim.x + threadIdx.x;
    int e = gid >> 4;
    if (e >= E) return;
    int src = ei[e], dst = ei[E + e];
    float T = __uint_as_float(state[2]);
    if (!(score[src] > T)) return;
    float g = gate[src];
    int d0 = (gid & 15) << 2;
    const float4 v = *(const float4*)(xtan + (size_t)src * 64 + d0);
    float* p = accg + (size_t)dst * 64 + d0;
    atomicAdd(p + 0, g * v.x); atomicAdd(p + 1, g * v.y);
    atomicAdd(p + 2, g * v.z); atomicAdd(p + 3, g * v.w);
}

// out = proj(expmap0(relu(proj(expmap0(x_tan + relu(acc_g))))))
__global__ void final_kernel(const float* __restrict__ xtan, const float* __restrict__ accg,
                             float* __restrict__ out, int N)
{
    int n = blockIdx.x * blockDim.x + threadIdx.x;
    if (n >= N) return;
    const float* xt = xtan + (size_t)n * 64;
    const float* ag = accg + (size_t)n * 64;
    float u2 = 0.f;
    for (int d = 0; d < 64; ++d) {
        float u = xt[d] + fmaxf(ag[d], 0.f);
        u2 = fmaf(u, u, u2);
    }
    float g1 = expmap0_proj_scale(u2);        // e = g1*u
    float r2 = 0.f;
    for (int d = 0; d < 64; ++d) {
        float u = xt[d] + fmaxf(ag[d], 0.f);
        float r = fmaxf(g1 * u, 0.f);         // relu
        r2 = fmaf(r, r, r2);
    }
    float g2 = expmap0_proj_scale(r2);
    float* o = out + (size_t)n * 64;
    for (int d = 0; d < 64; ++d) {
        float u = xt[d] + fmaxf(ag[d], 0.f);
        o[d] = g2 * fmaxf(g1 * u, 0.f);
    }
}

// ------------------------------------------------------------------- launcher
extern "C" void kernel_launch(void* const* d_in, const int* in_sizes, int n_in,
                              void* d_out, int out_size, void* d_ws, size_t ws_size,
                              hipStream_t stream) {
    (void)n_in; (void)out_size; (void)ws_size;
    const float* x    = (const float*)d_in[0];
    const int*   ei   = (const int*)d_in[1];
    const float* W    = (const float*)d_in[2];
    const float* bias = (const float*)d_in[3];
    const float* lw_w = (const float*)d_in[4];
    const float* lw_b = (const float*)d_in[5];
    int N = in_sizes[0] / 64;
    int E = in_sizes[1] / 2;
    float* outp = (float*)d_out;

    size_t ND = (size_t)N * 64;
    float* ws     = (float*)d_ws;
    float* xtan   = ws;                 // N*64
    float* neigh  = ws + ND;            // N*64 (zeroed)
    float* info   = ws + 2 * ND;        // N*64 (zeroed)
    float* selsum = ws + 3 * ND;        // N*64 (zeroed)
    float* accg   = ws + 4 * ND;        // N*64 (zeroed)
    float* deg    = ws + 5 * ND;        // N    (zeroed)
    float* dinv   = deg + N;            // N
    float* score  = dinv + N;           // N
    float* gate   = score + N;          // N
    float* hb     = gate + N;           // 128  (hb[64] = |hyp_bias|^2)
    unsigned* hist  = (unsigned*)(hb + 128);   // 256 (zeroed)
    unsigned* state = hist + 256;              // 8   (zeroed, then init)

    unsigned k = (unsigned)((double)N * 0.75); // RATIO

    // Zero everything from `neigh` through state (contiguous).
    size_t zeroFloats = 4 * ND + 4 * (size_t)N + 128 + 256 + 8;
    zeroFloats = (zeroFloats + 3) & ~(size_t)3;
    size_t n4 = zeroFloats / 4;
    zero_kernel<<<dim3((unsigned)((n4 + 255) / 256)), dim3(256), 0, stream>>>((float4*)neigh, n4);
    select_init_kernel<<<dim3(1), dim3(1), 0, stream>>>(state, k);

    bias_kernel<<<dim3(1), dim3(64), 0, stream>>>(bias, hb);
    hyplinear_kernel<<<dim3((unsigned)((N + 127) / 128)), dim3(256), 0, stream>>>(x, W, hb, xtan, N);

    unsigned gridE  = (unsigned)((E + 255) / 256);
    unsigned gridE16 = (unsigned)(((size_t)E * 16 + 255) / 256);
    unsigned gridN  = (unsigned)((N + 255) / 256);

    deg_kernel<<<dim3(gridE), dim3(256), 0, stream>>>(ei, deg, E);
    dinv_kernel<<<dim3(gridN), dim3(256), 0, stream>>>(deg, dinv, N);
    edge_passA<<<dim3(gridE16), dim3(256), 0, stream>>>(ei, xtan, dinv, neigh, info, E);
    score_kernel<<<dim3(gridN), dim3(256), 0, stream>>>(xtan, info, score, N);

    for (int pass = 0; pass < 4; ++pass) {
        hist_kernel<<<dim3(gridN), dim3(256), 0, stream>>>(score, hist, state, pass, N);
        scan_kernel<<<dim3(1), dim3(256), 0, stream>>>(hist, state, pass);
    }

    edge_passC<<<dim3(gridE16), dim3(256), 0, stream>>>(ei, xtan, score, state, selsum, E);
    gate_kernel<<<dim3(gridN), dim3(256), 0, stream>>>(selsum, neigh, lw_w, lw_b, gate, N);
    edge_passD<<<dim3(gridE16), dim3(256), 0, stream>>>(ei, xtan, score, gate, state, accg, E);
    final_kernel<<<dim3(gridN), dim3(256), 0, stream>>>(xtan, accg, outp, N);
}